// TemporalResonanceLayer_65377992180135
// MI455X (gfx1250) — compile-verified
//
#include <hip/hip_runtime.h>
#include <math.h>

// Shapes fixed by the reference: B=4, n=4096, d=1024, K=256.
#define BB     4
#define NN     4096
#define DD     1024
#define KK     256
#define MROWS  (BB * NN)          // 16384

#define AMP_MAX   3.0f
#define GATE_BIAS 0.6190392084062235f   // log(0.65/0.35)

typedef float v2f __attribute__((ext_vector_type(2)));
typedef float v8f __attribute__((ext_vector_type(8)));
typedef int   v2i __attribute__((ext_vector_type(2)));

__device__ __forceinline__ float sigmoid_f(float x) {
    return 1.0f / (1.0f + expf(-x));
}

// ---- CDNA5 async global->LDS copy (ASYNCcnt-tracked, no VGPR round trip) ----
__device__ __forceinline__ void async_copy_b64(float* lds_dst, const float* gsrc) {
#if __has_builtin(__builtin_amdgcn_global_load_async_to_lds_b64)
    typedef __attribute__((address_space(1))) v2i* gp_t;
    typedef __attribute__((address_space(3))) v2i* lp_t;
    float* g = const_cast<float*>(gsrc);
    __builtin_amdgcn_global_load_async_to_lds_b64(
        (gp_t)(void*)g, (lp_t)(void*)lds_dst, 0, 0);
#else
    asm volatile("global_load_async_to_lds_b64 %0, %1, off"
                 :: "v"((unsigned)(unsigned long long)lds_dst), "v"(gsrc)
                 : "memory");
#endif
}

__device__ __forceinline__ void wait_async0() {
#if __has_builtin(__builtin_amdgcn_s_wait_asynccnt)
    __builtin_amdgcn_s_wait_asynccnt(0);
#else
    asm volatile("s_wait_asynccnt 0x0" ::: "memory");
#endif
}

// -------------------------------------------------------------------------
// Kernel 1: raw = x @ W_proj^T + b_proj, fused with the oscillator epilogue.
// Block = 256 threads (8 waves) owns 256 rows and one 16-col k-tile (same
// columns in all 4 chunks of the 4K output). The shared B panel (64 cols x
// 64 k) is staged into LDS with CDNA5 async-to-LDS loads and reused by all
// 8 waves. Each wave computes 32 rows x 16 cols x 4 chunks = 8 accumulators.
// Grid: (MROWS/256) * (KK/16) = 64*16 = 1024 blocks.
// -------------------------------------------------------------------------
#define KSLAB 64
#define BPITCH 66   // floats; pad to spread LDS banks, keeps 8B alignment

__global__ __launch_bounds__(256)
void proj_osc_kernel(const float* __restrict__ x,
                     const float* __restrict__ Wp,
                     const float* __restrict__ bp,
                     float* __restrict__ alpha_buf,
                     float* __restrict__ dr_buf,
                     float* __restrict__ di_buf,
                     float* __restrict__ ang_buf)
{
    __shared__ float bsl[64][BPITCH];   // [chunk*16+col][k] B slab

    const int tid  = threadIdx.x;
    const int lane = tid & 31;
    const int wave = tid >> 5;          // 0..7
    const int bm   = blockIdx.x >> 4;   // 0..63  (256-row macro tile)
    const int kt   = blockIdx.x & 15;   // 0..15  (16-col k tile)
    const int kb   = kt * 16;
    const int lo = lane & 15;
    const int hi = lane >> 4;           // 0 or 1 (half-wave)

    const int rowBase = bm * 256 + wave * 32;

    // A fragments (16x4 f32): lanes 0-15 hold K={0,1}, lanes 16-31 K={2,3}
    const float* arow0 = x + (size_t)(rowBase + lo) * DD + 2 * hi;
    const float* arow1 = arow0 + (size_t)16 * DD;

    // Async staging pattern: 2048 b64 transfers per slab, 8 per thread.
    // f = it*256+tid: col = f>>5 (0..63), kh = f&31 -> k offset 2*kh.
    v8f acc[2][4] = {};

    for (int slab = 0; slab < DD; slab += KSLAB) {
        __syncthreads();   // previous slab fully consumed before overwrite
#pragma unroll
        for (int it = 0; it < 8; ++it) {
            const int f   = it * 256 + tid;
            const int col = f >> 5;          // 0..63 = chunk*16 + colInTile
            const int kh  = f & 31;          // -> floats 2*kh, 2*kh+1
            const int colGlobal = (col >> 4) * KK + kb + (col & 15);
            async_copy_b64(&bsl[col][2 * kh],
                           Wp + (size_t)colGlobal * DD + slab + 2 * kh);
        }
        wait_async0();
        __syncthreads();

#pragma unroll 4
        for (int kk = 0; kk < KSLAB; kk += 4) {
            v2f a0 = *(const v2f*)(arow0 + slab + kk);
            v2f a1 = *(const v2f*)(arow1 + slab + kk);
#pragma unroll
            for (int c = 0; c < 4; ++c) {
                v2f b = *(const v2f*)&bsl[c * 16 + lo][kk + 2 * hi];
                acc[0][c] = __builtin_amdgcn_wmma_f32_16x16x4_f32(
                    false, a0, false, b, (short)0, acc[0][c], false, false);
                acc[1][c] = __builtin_amdgcn_wmma_f32_16x16x4_f32(
                    false, a1, false, b, (short)0, acc[1][c], false, false);
            }
        }
    }

    const float bias0 = bp[0 * KK + kb + lo];
    const float bias1 = bp[1 * KK + kb + lo];
    const float bias2 = bp[2 * KK + kb + lo];
    const float bias3 = bp[3 * KK + kb + lo];

    // C/D layout: VGPR i -> row M = i (lanes 0-15) or 8+i (lanes 16-31); N = lane%16
#pragma unroll
    for (int ms = 0; ms < 2; ++ms) {
#pragma unroll
        for (int i = 0; i < 8; ++i) {
            const int grow = rowBase + ms * 16 + i + hi * 8;
            const int nidx = grow & (NN - 1);

            const float a_raw = acc[ms][0][i] + bias0;
            const float w_raw = acc[ms][1][i] + bias1;
            const float p_raw = acc[ms][2][i] + bias2;
            const float g_raw = acc[ms][3][i] + bias3;

            const float A     = AMP_MAX * sigmoid_f(a_raw);
            const float omega = (w_raw > 20.0f) ? w_raw : log1pf(expf(w_raw));
            const float alpha = sigmoid_f(g_raw + GATE_BIAS);
            const float pos   = log1pf((float)nidx);
            const float ang   = fmaf(omega, pos, p_raw);

            float s, c;
            __sincosf(ang, &s, &c);
            const float oma = (1.0f - alpha) * A;

            const size_t idx = (size_t)grow * KK + kb + lo;
            alpha_buf[idx] = alpha;
            dr_buf[idx]    = oma * c;
            di_buf[idx]    = oma * s;
            ang_buf[idx]   = ang;
        }
    }
}

// -------------------------------------------------------------------------
// Kernel 2: work-efficient scan over time + normalize + demodulate -> rho.
// Block = 1024 threads = (16 channels) x (64 time chunks of 64 steps).
// rho_buf aliases alpha_buf (single-owner indices, read-before-write),
// so those two params are NOT __restrict__.
// -------------------------------------------------------------------------
#define SCAN_TK     16
#define SCAN_CHUNKS 64
#define SCAN_CH     (NN / SCAN_CHUNKS)   // 64

__global__ __launch_bounds__(1024)
void scan_demod_kernel(const float* alpha_buf,
                       const float* __restrict__ dr_buf,
                       const float* __restrict__ di_buf,
                       const float* __restrict__ ang_buf,
                       float* rho_buf)
{
    const int b  = blockIdx.x >> 4;            // 0..3
    const int kb = (blockIdx.x & 15) * SCAN_TK;
    const int tk = threadIdx.x & (SCAN_TK - 1);
    const int tc = threadIdx.x / SCAN_TK;      // 0..63 time chunk
    const int k  = kb + tk;
    const size_t chan_base = (size_t)b * NN * KK + k;
    const int n0 = tc * SCAN_CH;

    // Pass 1: per-chunk summary of r_t = a_t * r_{t-1} + b_t  (complex b)
    float pa = 1.0f, pr = 0.0f, pi = 0.0f;
#pragma unroll 4
    for (int j = 0; j < SCAN_CH; ++j) {
        const size_t idx = chan_base + (size_t)(n0 + j) * KK;
        const float a = alpha_buf[idx];
        pr = fmaf(a, pr, dr_buf[idx]);
        pi = fmaf(a, pi, di_buf[idx]);
        pa *= a;
    }

    __shared__ float sa[SCAN_CHUNKS][SCAN_TK];
    __shared__ float sr[SCAN_CHUNKS][SCAN_TK];
    __shared__ float si[SCAN_CHUNKS][SCAN_TK];
    sa[tc][tk] = pa; sr[tc][tk] = pr; si[tc][tk] = pi;
    __syncthreads();

    // Exclusive scan across chunks (serial, 16 threads x 64 tiny steps)
    if (tc == 0) {
        float er = 0.0f, ei = 0.0f;
        for (int c2 = 0; c2 < SCAN_CHUNKS; ++c2) {
            const float ta = sa[c2][tk];
            const float tr = sr[c2][tk];
            const float ti = si[c2][tk];
            sr[c2][tk] = er;
            si[c2][tk] = ei;
            er = fmaf(ta, er, tr);
            ei = fmaf(ta, ei, ti);
        }
    }
    __syncthreads();

    // Pass 2: seed with exclusive prefix, replay recurrence, fuse demod
    float rr = sr[tc][tk];
    float ri = si[tc][tk];
#pragma unroll 4
    for (int j = 0; j < SCAN_CH; ++j) {
        const size_t idx = chan_base + (size_t)(n0 + j) * KK;
        const float a = alpha_buf[idx];
        rr = fmaf(a, rr, dr_buf[idx]);
        ri = fmaf(a, ri, di_buf[idx]);
        const float ang = ang_buf[idx];
        float s, c;
        __sincosf(ang, &s, &c);
        const float scale = fmaxf(fmaxf(fabsf(rr), fabsf(ri)), 1.0f);
        rho_buf[idx] = (rr * c + ri * s) / scale;   // written AFTER alpha read
    }
}

// -------------------------------------------------------------------------
// Kernel 3: out = res_scale * rho @ W_res^T  (M=16384, N=1024, K=256), fp32
// WMMA. Each wave computes a 32x32 tile (2x2 subtiles, 4 accumulators) to
// halve L2 fragment traffic. Operands (rho 16MB + W_res 1MB) are L2-hot.
// Grid: (16384/32)*(1024/32) = 16384 waves -> 2048 blocks.
// -------------------------------------------------------------------------
__global__ __launch_bounds__(256)
void resout_kernel(const float* __restrict__ rho,
                   const float* __restrict__ Wr,
                   const float* __restrict__ res_scale,
                   float* __restrict__ out)
{
    const int lane = threadIdx.x & 31;
    const int wid  = (blockIdx.x * blockDim.x + threadIdx.x) >> 5;
    const int mt   = wid >> 5;          // 0..511
    const int nt   = wid & 31;          // 0..31
    const int mbase = mt * 32;
    const int nb    = nt * 32;
    const int lo = lane & 15;
    const int hi = lane >> 4;

    const float* arow0 = rho + (size_t)(mbase + lo) * KK + 2 * hi;
    const float* arow1 = arow0 + (size_t)16 * KK;
    const float* brow0 = Wr  + (size_t)(nb + lo)    * KK + 2 * hi;
    const float* brow1 = brow0 + (size_t)16 * KK;

    v8f acc[2][2] = {};
#pragma unroll 4
    for (int kk = 0; kk < KK; kk += 4) {
        v2f a0 = *(const v2f*)(arow0 + kk);
        v2f a1 = *(const v2f*)(arow1 + kk);
        v2f b0 = *(const v2f*)(brow0 + kk);
        v2f b1 = *(const v2f*)(brow1 + kk);
        acc[0][0] = __builtin_amdgcn_wmma_f32_16x16x4_f32(false, a0, false, b0, (short)0, acc[0][0], false, false);
        acc[0][1] = __builtin_amdgcn_wmma_f32_16x16x4_f32(false, a0, false, b1, (short)0, acc[0][1], false, false);
        acc[1][0] = __builtin_amdgcn_wmma_f32_16x16x4_f32(false, a1, false, b0, (short)0, acc[1][0], false, false);
        acc[1][1] = __builtin_amdgcn_wmma_f32_16x16x4_f32(false, a1, false, b1, (short)0, acc[1][1], false, false);
    }

    const float rs = res_scale[0];
#pragma unroll
    for (int mi = 0; mi < 2; ++mi) {
#pragma unroll
        for (int ni = 0; ni < 2; ++ni) {
#pragma unroll
            for (int i = 0; i < 8; ++i) {
                const int grow = mbase + mi * 16 + i + hi * 8;
                out[(size_t)grow * DD + nb + ni * 16 + lo] = rs * acc[mi][ni][i];
            }
        }
    }
}

// -------------------------------------------------------------------------
extern "C" void kernel_launch(void* const* d_in, const int* in_sizes, int n_in,
                              void* d_out, int out_size, void* d_ws, size_t ws_size,
                              hipStream_t stream) {
    const float* x         = (const float*)d_in[0];  // (B, n, d)
    const float* W_proj    = (const float*)d_in[1];  // (4K, d)
    const float* b_proj    = (const float*)d_in[2];  // (4K,)
    const float* W_res     = (const float*)d_in[3];  // (d, K)
    const float* res_scale = (const float*)d_in[4];  // scalar

    float* out = (float*)d_out;

    // Workspace: 4 arrays of B*n*K floats (16 MB each); rho aliases alpha.
    const size_t elems = (size_t)BB * NN * KK;       // 4,194,304
    float* alpha_buf = (float*)d_ws;
    float* dr_buf    = alpha_buf + elems;
    float* di_buf    = dr_buf + elems;
    float* ang_buf   = di_buf + elems;
    float* rho_buf   = alpha_buf;                    // safe alias (see kernel 2)

    // Kernel 1: (16384/256) * 16 = 1024 blocks of 256 threads
    proj_osc_kernel<<<(MROWS / 256) * (KK / 16), 256, 0, stream>>>(
        x, W_proj, b_proj, alpha_buf, dr_buf, di_buf, ang_buf);

    // Kernel 2: B * (K/16) = 64 blocks of 1024 threads
    scan_demod_kernel<<<BB * (KK / SCAN_TK), 1024, 0, stream>>>(
        alpha_buf, dr_buf, di_buf, ang_buf, rho_buf);

    // Kernel 3: 16384 waves / 8 per block = 2048 blocks
    resout_kernel<<<(MROWS / 32) * (DD / 32) / 8, 256, 0, stream>>>(
        rho_buf, W_res, res_scale, out);
}